// GCNGraphClassifier_56882546868465
// MI455X (gfx1250) — compile-verified
//
#include <hip/hip_runtime.h>
#include <hip/hip_bf16.h>

#define NN   50000
#define EE   800000
#define GG   1000
#define HH   128
#define LL   3
#define CC   6
#define BN_EPS 1e-5f

typedef float v2f __attribute__((ext_vector_type(2)));
typedef float v8f __attribute__((ext_vector_type(8)));

// ---------------- generic zero ----------------
__global__ void k_zero(float* p, int n) {
    int i = blockIdx.x * blockDim.x + threadIdx.x;
    if (i < n) p[i] = 0.0f;
}

// ---------------- degree: atomic count over dst ----------------
__global__ void k_deg(const int* dst, float* deg, int e) {
    int i = blockIdx.x * blockDim.x + threadIdx.x;
    if (i < e) atomicAdd(&deg[dst[i]], 1.0f);
}

// dis[n] = rsqrt(deg[n] + 1)   (in place over deg buffer)
__global__ void k_dis(float* deg_dis, int n) {
    int i = blockIdx.x * blockDim.x + threadIdx.x;
    if (i < n) deg_dis[i] = rsqrtf(deg_dis[i] + 1.0f);
}

// ---------------- WMMA fp32 GEMM:  D[M,128] = A[M,K] @ B[K,128] (+bias, relu) --
// One wave owns a 16x128 output strip: 8 accumulator tiles, A-fragment reused
// across 8 independent V_WMMA_F32_16X16X4_F32 per K-step.
template <bool BOUNDED>
__global__ void k_gemm_wmma(const float* __restrict__ A, const float* __restrict__ B,
                            const float* __restrict__ bias, float* __restrict__ D,
                            int M, int K, int relu) {
    int mtiles = (M + 15) >> 4;
    int mt = blockIdx.x * (blockDim.x >> 5) + (threadIdx.x >> 5);
    if (mt >= mtiles) return;                      // uniform per wave

    int lane = threadIdx.x & 31;
    int m0 = mt << 4;
    int lrow = m0 + (lane & 15);                   // A row for this lane
    int koff = (lane >> 4) << 1;                   // 0 for lanes 0-15, 2 for 16-31
    int ncol = lane & 15;                          // column within an N tile

    int arow = lrow;
    bool rok = true;
    if (BOUNDED) { rok = (lrow < M); arow = rok ? lrow : (M - 1); }

    v8f acc[8];
    #pragma unroll
    for (int nt = 0; nt < 8; ++nt)
        acc[nt] = (v8f){0.f, 0.f, 0.f, 0.f, 0.f, 0.f, 0.f, 0.f};

    #pragma unroll 2
    for (int k0 = 0; k0 < K; k0 += 4) {
        int kk = k0 + koff;
        v2f av = *(const v2f*)(A + (size_t)arow * K + kk);   // 8B-aligned
        if (BOUNDED && !rok) { av.x = 0.f; av.y = 0.f; }
        const float* b0p = B + (size_t)kk * HH + ncol;       // K row kk
        const float* b1p = b0p + HH;                         // K row kk+1
        #pragma unroll
        for (int nt = 0; nt < 8; ++nt) {
            v2f bv = { b0p[nt * 16], b1p[nt * 16] };
            // (neg_a, A, neg_b, B, c_mod, C, reuse_a, reuse_b)
            acc[nt] = __builtin_amdgcn_wmma_f32_16x16x4_f32(false, av, false, bv,
                                                            (short)0, acc[nt],
                                                            false, false);
        }
    }

    int rbase = m0 + ((lane >> 4) << 3);           // +8 for upper half-lanes
    #pragma unroll
    for (int nt = 0; nt < 8; ++nt) {
        int col = nt * 16 + ncol;
        float bb = bias ? bias[col] : 0.0f;
        #pragma unroll
        for (int r = 0; r < 8; ++r) {
            int row = rbase + r;
            if (!BOUNDED || row < M) {
                float v = acc[nt][r] + bb;
                if (relu) v = fmaxf(v, 0.0f);
                D[(size_t)row * HH + col] = v;
            }
        }
    }
}

// ---------------- agg init: agg = h * dis^2 + conv_b ----------------
__global__ void k_init_agg(const float* __restrict__ h, const float* __restrict__ dis,
                           const float* __restrict__ cb, float* __restrict__ agg, int total) {
    int i = blockIdx.x * blockDim.x + threadIdx.x;
    if (i < total) {
        int n = i >> 7;
        float s = dis[n];
        agg[i] = h[i] * s * s + cb[i & 127];
    }
}

// ---------------- edge scatter: warp per edge, float4 per lane ----------------
__global__ void k_scatter(const int* __restrict__ src, const int* __restrict__ dst,
                          const float* __restrict__ dis, const float* __restrict__ h,
                          float* __restrict__ agg, int e) {
    int w = (blockIdx.x * blockDim.x + threadIdx.x) >> 5;   // edge id (uniform per wave)
    int lane = threadIdx.x & 31;
    if (w >= e) return;
    int s = src[w], d = dst[w];
    float norm = dis[s] * dis[d];
    const float4* hp = (const float4*)(h + (size_t)s * HH);
    float4 v = hp[lane];
    float* ap = agg + (size_t)d * HH + lane * 4;
    atomicAdd(ap + 0, v.x * norm);
    atomicAdd(ap + 1, v.y * norm);
    atomicAdd(ap + 2, v.z * norm);
    atomicAdd(ap + 3, v.w * norm);
}

// ---------------- BN partial sums: thread = channel, block = row chunk ----------
__global__ void k_bn_partial(const float* __restrict__ a, float* __restrict__ stats, int n) {
    int ch = threadIdx.x;                     // 0..127
    int r0 = blockIdx.x * 256;
    float s = 0.f, s2 = 0.f;
    for (int r = 0; r < 256; ++r) {
        int row = r0 + r;
        if (row < n) {
            float v = a[row * HH + ch];
            s += v; s2 += v * v;
        }
    }
    atomicAdd(&stats[ch], s);
    atomicAdd(&stats[HH + ch], s2);
}

// scale/shift: y = x*scale + shift, scale = g*rsqrt(var+eps), shift = b - mu*scale
__global__ void k_bn_final(const float* __restrict__ stats, const float* __restrict__ g,
                           const float* __restrict__ b, float* __restrict__ scsh, int n) {
    int ch = threadIdx.x;
    if (ch < HH) {
        float inv_n = 1.0f / (float)n;
        float mu  = stats[ch] * inv_n;
        float var = stats[HH + ch] * inv_n - mu * mu;
        float sc  = g[ch] * rsqrtf(var + BN_EPS);
        scsh[ch]       = sc;
        scsh[HH + ch]  = b[ch] - mu * sc;
    }
}

__global__ void k_bn_apply(const float* __restrict__ a, const float* __restrict__ scsh,
                           float* __restrict__ out, int total) {
    int i = blockIdx.x * blockDim.x + threadIdx.x;
    if (i < total) {
        int ch = i & 127;
        out[i] = fmaxf(a[i] * scsh[ch] + scsh[HH + ch], 0.0f);
    }
}

// ---------------- pooling ----------------
__global__ void k_pool_scatter(const float* __restrict__ feat, const int* __restrict__ batch,
                               float* __restrict__ psum, float* __restrict__ cnt) {
    int n = blockIdx.x;
    int g = batch[n];
    atomicAdd(&psum[g * HH + threadIdx.x], feat[n * HH + threadIdx.x]);
    if (threadIdx.x == 0) atomicAdd(&cnt[g], 1.0f);
}

__global__ void k_pool_final(float* __restrict__ psum, const float* __restrict__ cnt, int total) {
    int i = blockIdx.x * blockDim.x + threadIdx.x;
    if (i < total) psum[i] = psum[i] / fmaxf(cnt[i >> 7], 1.0f);
}

// ---------------- final classifier: out[g,c] = t[g,:] . W2[:,c] + b2[c] --------
__global__ void k_mlp2(const float* __restrict__ t, const float* __restrict__ w2,
                       const float* __restrict__ b2, float* __restrict__ out, int g_total) {
    int i = blockIdx.x * blockDim.x + threadIdx.x;
    if (i < g_total * CC) {
        int g = i / CC, c = i % CC;
        float acc = b2[c];
        #pragma unroll 4
        for (int k = 0; k < HH; ++k)
            acc += t[g * HH + k] * w2[k * CC + c];
        out[i] = acc;
    }
}

extern "C" void kernel_launch(void* const* d_in, const int* in_sizes, int n_in,
                              void* d_out, int out_size, void* d_ws, size_t ws_size,
                              hipStream_t stream) {
    const float* x      = (const float*)d_in[0];
    const int*   eidx   = (const int*)d_in[1];     // [2,E]: src then dst
    const int*   batch  = (const int*)d_in[2];
    const float* conv_w = (const float*)d_in[3];   // [L,H,H]
    const float* conv_b = (const float*)d_in[4];   // [L,H]
    const float* bn_g   = (const float*)d_in[5];
    const float* bn_b   = (const float*)d_in[6];
    const float* mlp_w1 = (const float*)d_in[7];
    const float* mlp_b1 = (const float*)d_in[8];
    const float* mlp_w2 = (const float*)d_in[9];
    const float* mlp_b2 = (const float*)d_in[10];
    float* out = (float*)d_out;

    const int* src = eidx;
    const int* dst = eidx + EE;

    // workspace layout (floats)
    const size_t NH = (size_t)NN * HH;
    float* W    = (float*)d_ws;
    float* dis  = W;                        // N
    float* h    = dis + NN;                 // N*H
    float* agg  = h   + NH;                 // N*H
    float* feat = agg + NH;                 // N*H
    float* stats= feat+ NH;                 // 2*H
    float* scsh = stats + 2 * HH;           // 2*H
    float* psum = scsh  + 2 * HH;           // G*H
    float* cnt  = psum  + (size_t)GG * HH;  // G   (contiguous after psum)
    float* tbuf = cnt   + GG;               // G*H

    // --- degree / normalization ---
    k_zero<<<(NN + 255) / 256, 256, 0, stream>>>(dis, NN);
    k_deg<<<(EE + 255) / 256, 256, 0, stream>>>(dst, dis, EE);
    k_dis<<<(NN + 255) / 256, 256, 0, stream>>>(dis, NN);

    const int total = NN * HH;
    // GEMM grid: 4 waves per 128-thread block, one 16x128 strip per wave
    const int conv_mtiles = NN / 16;                       // 3125 (exact)
    const int conv_grid   = (conv_mtiles + 3) / 4;

    for (int l = 0; l < LL; ++l) {
        const float* fin = (l == 0) ? x : feat;
        // h = fin @ conv_w[l]   (M multiple of 16 -> unbounded fast path)
        k_gemm_wmma<false><<<conv_grid, 128, 0, stream>>>(fin, conv_w + (size_t)l * HH * HH,
                                                          nullptr, h, NN, HH, 0);
        // agg = h*dis^2 + b
        k_init_agg<<<(total + 255) / 256, 256, 0, stream>>>(h, dis, conv_b + l * HH, agg, total);
        // agg[dst] += h[src] * dis[src]*dis[dst]
        k_scatter<<<(EE * 32 + 255) / 256, 256, 0, stream>>>(src, dst, dis, h, agg, EE);
        // batchnorm + relu -> feat
        k_zero<<<1, 256, 0, stream>>>(stats, 2 * HH);
        k_bn_partial<<<(NN + 255) / 256, 128, 0, stream>>>(agg, stats, NN);
        k_bn_final<<<1, 128, 0, stream>>>(stats, bn_g + l * HH, bn_b + l * HH, scsh, NN);
        k_bn_apply<<<(total + 255) / 256, 256, 0, stream>>>(agg, scsh, feat, total);
    }

    // --- global mean pool ---
    k_zero<<<(GG * HH + GG + 255) / 256, 256, 0, stream>>>(psum, GG * HH + GG);
    k_pool_scatter<<<NN, 128, 0, stream>>>(feat, batch, psum, cnt);
    k_pool_final<<<(GG * HH + 255) / 256, 256, 0, stream>>>(psum, cnt, GG * HH);

    // --- MLP head ---
    const int m1_mtiles = (GG + 15) / 16;                  // 63 (needs bounds)
    k_gemm_wmma<true><<<(m1_mtiles + 3) / 4, 128, 0, stream>>>(psum, mlp_w1, mlp_b1, tbuf,
                                                               GG, HH, 1);
    k_mlp2<<<(GG * CC + 255) / 256, 256, 0, stream>>>(tbuf, mlp_w2, mlp_b2, out, GG);
}